// AudioDecoderWrapper_82051055223101
// MI455X (gfx1250) — compile-verified
//
#include <hip/hip_runtime.h>
#include <math.h>

// Model dims
#define L 16
#define D 1024
#define H 16
#define HD 64
#define CC 4096
#define FF 4096
#define NQ 16
#define CB 2048
#define QD 512
#define KCHUNK 128   // split-K chunk for WMMA GEMV

typedef float v2f __attribute__((ext_vector_type(2)));
typedef float v4f __attribute__((ext_vector_type(4)));
typedef float v8f __attribute__((ext_vector_type(8)));

// ---------------------------------------------------------------------------
// Embedding: emb[qd] = sum_nq codebooks[nq, codes[nq], qd]
// ---------------------------------------------------------------------------
__global__ void embed_kernel(const int* __restrict__ codes,
                             const float* __restrict__ codebooks,
                             float* __restrict__ emb) {
    int qd = blockIdx.x * blockDim.x + threadIdx.x;
    if (qd >= QD) return;
    float s = 0.f;
    for (int n = 0; n < NQ; ++n) {
        int c = codes[n];
        s += codebooks[((size_t)n * CB + c) * QD + qd];
    }
    emb[qd] = s;
}

// ---------------------------------------------------------------------------
// y[i] = bias ? bias[i] : 0
// ---------------------------------------------------------------------------
__global__ void init_kernel(float* __restrict__ y, const float* __restrict__ bias, int n) {
    int i = blockIdx.x * blockDim.x + threadIdx.x;
    if (i < n) y[i] = bias ? bias[i] : 0.f;
}

// ---------------------------------------------------------------------------
// GEMV y[n] += sum_k x[k] * W[k,n] using V_WMMA_F32_16X16X4_F32.
// A row M=0 = x chunk (other rows zeroed via a branch-free 0/1 lane mask).
// B = 4x16 fp32 weight tile, streamed with non-temporal loads (single-use).
// D row 0 (lanes 0..15, VGPR0) = 16 partial outputs -> atomicAdd (split-K).
// 32-bit A 16x4 layout: VGPR0 = K=0 (lanes 0-15) / K=2 (lanes 16-31); VGPR1 = K=1 / K=3.
// ---------------------------------------------------------------------------
__global__ void wmma_gemv_kernel(const float* __restrict__ x,
                                 const float* __restrict__ W,
                                 float* __restrict__ y, int K, int N) {
    int wave = (int)((blockIdx.x * blockDim.x + threadIdx.x) >> 5);
    int lane = threadIdx.x & 31;
    int ntiles = N >> 4;
    int ksplit = K / KCHUNK;
    if (wave >= ntiles * ksplit) return;
    int n0 = (wave % ntiles) << 4;
    int k0 = (wave / ntiles) * KCHUNK;
    int halfsel = (lane >> 4) << 1;          // 0 for lanes 0-15, 2 for lanes 16-31
    int col = n0 + (lane & 15);
    float xmask = ((lane & 15) == 0) ? 1.f : 0.f;

    const float* Wp = W + (size_t)(k0 + halfsel) * N + col;
    const float* xp = x + k0 + halfsel;

    v8f c = {0.f, 0.f, 0.f, 0.f, 0.f, 0.f, 0.f, 0.f};
#pragma unroll 4
    for (int it = 0; it < KCHUNK / 4; ++it) {
        v2f xv = *(const v2f*)xp;            // one b64 load, broadcast per half-wave
        v2f a;
        a.x = xmask * xv.x;
        a.y = xmask * xv.y;
        v2f b;
        b.x = __builtin_nontemporal_load(Wp);
        b.y = __builtin_nontemporal_load(Wp + N);
        c = __builtin_amdgcn_wmma_f32_16x16x4_f32(false, a, false, b, (short)0, c, false, false);
        xp += 4;
        Wp += (size_t)4 * N;
    }
    if (lane < 16) atomicAdd(&y[n0 + lane], c[0]);
}

// ---------------------------------------------------------------------------
// LayerNorm over D, one block of 256 threads.
// ---------------------------------------------------------------------------
__global__ void ln_kernel(const float* __restrict__ x, const float* __restrict__ g,
                          const float* __restrict__ b, float* __restrict__ out) {
    __shared__ float rbuf[256];
    int tid = threadIdx.x;
    float s = 0.f;
    for (int i = tid; i < D; i += 256) s += x[i];
    rbuf[tid] = s; __syncthreads();
    for (int st = 128; st > 0; st >>= 1) { if (tid < st) rbuf[tid] += rbuf[tid + st]; __syncthreads(); }
    float mean = rbuf[0] * (1.f / D);
    __syncthreads();
    float v = 0.f;
    for (int i = tid; i < D; i += 256) { float d = x[i] - mean; v += d * d; }
    rbuf[tid] = v; __syncthreads();
    for (int st = 128; st > 0; st >>= 1) { if (tid < st) rbuf[tid] += rbuf[tid + st]; __syncthreads(); }
    float inv = rsqrtf(rbuf[0] * (1.f / D) + 1e-5f);
    for (int i = tid; i < D; i += 256)
        out[i] = (x[i] - mean) * inv * g[i] + b[i];
}

// ---------------------------------------------------------------------------
// RoPE on q and k slices of qkv. 512 threads: h = tid/32, i = tid%32.
// ---------------------------------------------------------------------------
__global__ void rope_kernel(const float* __restrict__ qkv, const int* __restrict__ offsets,
                            int layer, float* __restrict__ qout, float* __restrict__ kout) {
    int tid = blockIdx.x * blockDim.x + threadIdx.x;
    if (tid >= H * 32) return;
    int h = tid >> 5, i = tid & 31;
    float pos = (float)offsets[layer];
    float inv = powf(10000.f, -(float)i / 32.f);
    float ang = pos * inv;
    float cs = cosf(ang), sn = sinf(ang);
    const float* qh = qkv + h * HD;
    const float* kh = qkv + D + h * HD;
    qout[h * HD + i]      = qh[i] * cs - qh[i + 32] * sn;
    qout[h * HD + i + 32] = qh[i + 32] * cs + qh[i] * sn;
    kout[h * HD + i]      = kh[i] * cs - kh[i + 32] * sn;
    kout[h * HD + i + 32] = kh[i + 32] * cs + kh[i] * sn;
}

// ---------------------------------------------------------------------------
// Attention for one layer: one block per head, 256 threads (8 waves).
// Phase 1: scores via chained V_WMMA_F32_16X16X4_F32 (16 keys per tile).
// Phase 2: softmax in LDS.  Phase 3: attn . V with coalesced float2 FMAs.
// ---------------------------------------------------------------------------
__global__ void attn_kernel(const float* __restrict__ q, const float* __restrict__ knew,
                            const float* __restrict__ qkv,
                            const float* __restrict__ cached_k, const float* __restrict__ cached_v,
                            const int* __restrict__ cached_pos, const int* __restrict__ offsets,
                            int layer, float* __restrict__ attnout) {
    __shared__ float sc[CC + 8];
    __shared__ float rbuf[256];
    __shared__ float red[8 * HD];
    int h = blockIdx.x;
    int tid = threadIdx.x;
    int lane = tid & 31;
    int wave = tid >> 5;
    const float* Kc = cached_k + ((size_t)layer * H + h) * (size_t)CC * HD;
    const float* Vc = cached_v + ((size_t)layer * H + h) * (size_t)CC * HD;
    const int* posk = cached_pos + layer * CC;
    int pos_q = offsets[layer];
    const float* qh = q + h * HD;
    int halfsel = (lane >> 4) << 1;
    int m = lane & 15;

    // Phase 1: scores over the 4096 cached keys, 16-key tiles per wave
    for (int t = wave; t < CC / 16; t += 8) {
        int j0 = t * 16;
        const float* Krow = Kc + (size_t)(j0 + m) * HD + halfsel;
        const float* qp = qh + halfsel;
        v8f c = {0.f, 0.f, 0.f, 0.f, 0.f, 0.f, 0.f, 0.f};
#pragma unroll
        for (int kk = 0; kk < HD; kk += 4) {
            v2f a = *(const v2f*)(Krow + kk);   // b64 load of K row chunk
            v2f b = *(const v2f*)(qp + kk);     // b64 load of q chunk (broadcast)
            c = __builtin_amdgcn_wmma_f32_16x16x4_f32(false, a, false, b, (short)0, c, false, false);
        }
        if (lane == 0 || lane == 16) {
            int mbase = j0 + ((lane == 16) ? 8 : 0);
            for (int r = 0; r < 8; ++r) {
                int j = mbase + r;
                int pk = posk[j];
                int dq = pos_q - pk;
                bool valid = (dq >= 0) && (dq < CC) && (pk >= 0);
                sc[j] = valid ? (c[r] * 0.125f) : -1e9f;   // 1/sqrt(64)
            }
        }
    }
    // New key (delta == 0, always valid)
    if (tid == 0) {
        float s = 0.f;
        const float* kn = knew + h * HD;
        for (int d = 0; d < HD; ++d) s += qh[d] * kn[d];
        sc[CC] = s * 0.125f;
    }
    __syncthreads();

    // Phase 2: softmax (unnormalized probs + denom)
    float mx = -1e30f;
    for (int j = tid; j <= CC; j += 256) mx = fmaxf(mx, sc[j]);
    rbuf[tid] = mx; __syncthreads();
    for (int st = 128; st > 0; st >>= 1) { if (tid < st) rbuf[tid] = fmaxf(rbuf[tid], rbuf[tid + st]); __syncthreads(); }
    mx = rbuf[0]; __syncthreads();
    float ls = 0.f;
    for (int j = tid; j <= CC; j += 256) { float p = __expf(sc[j] - mx); sc[j] = p; ls += p; }
    rbuf[tid] = ls; __syncthreads();
    for (int st = 128; st > 0; st >>= 1) { if (tid < st) rbuf[tid] += rbuf[tid + st]; __syncthreads(); }
    float denom = rbuf[0];

    // Phase 3: out[hd] = sum_j p[j] * V[j][hd]
    int hd2 = lane * 2;
    float accx = 0.f, accy = 0.f;
    for (int j = wave; j < CC; j += 8) {
        float p = sc[j];
        v2f v2 = *(const v2f*)(Vc + (size_t)j * HD + hd2);
        accx += p * v2.x; accy += p * v2.y;
    }
    if (wave == 0) {
        float p = sc[CC];
        const float* vn = qkv + 2 * D + h * HD;
        accx += p * vn[hd2]; accy += p * vn[hd2 + 1];
    }
    red[wave * HD + hd2] = accx; red[wave * HD + hd2 + 1] = accy;
    __syncthreads();
    if (tid < HD) {
        float s = 0.f;
        for (int w = 0; w < 8; ++w) s += red[w * HD + tid];
        attnout[h * HD + tid] = s / denom;
    }
}

// ---------------------------------------------------------------------------
// x[i] += ls[i] * t[i]
// ---------------------------------------------------------------------------
__global__ void resid_kernel(float* __restrict__ x, const float* __restrict__ ls,
                             const float* __restrict__ t, int n) {
    int i = blockIdx.x * blockDim.x + threadIdx.x;
    if (i < n) x[i] += ls[i] * t[i];
}

// exact gelu: 0.5*x*(1+erf(x/sqrt(2)))
__global__ void gelu_kernel(float* __restrict__ y, int n) {
    int i = blockIdx.x * blockDim.x + threadIdx.x;
    if (i < n) { float v = y[i]; y[i] = 0.5f * v * (1.f + erff(v * 0.70710678118654752f)); }
}

// ---------------------------------------------------------------------------
// KV-cache shift: dst[l,h,c,:] = src[l,h,c+1,:] for c < C-1.
// 2D grid: blockIdx.y = (l,h) pair -> no integer division in the hot path.
// Non-temporal b128 reads (clang ext-vector type), non-temporal b32 stores
// (d_out region is only 4-byte aligned).  Dominant 1 GB read+write stream.
// ---------------------------------------------------------------------------
__global__ void shift_cache_kernel(float* __restrict__ dst, const float* __restrict__ src) {
    const int per = (CC - 1) * HD;                        // floats per (l,h)
    size_t lh = blockIdx.y;
    int r = (blockIdx.x * blockDim.x + threadIdx.x) * 4;
    if (r >= per) return;
    const v4f* s = (const v4f*)(src + lh * (size_t)CC * HD + HD + r);
    float* d = dst + lh * (size_t)CC * HD + r;
    v4f v = __builtin_nontemporal_load(s);
    __builtin_nontemporal_store(v.x, d + 0);
    __builtin_nontemporal_store(v.y, d + 1);
    __builtin_nontemporal_store(v.z, d + 2);
    __builtin_nontemporal_store(v.w, d + 3);
}

__global__ void shift_pos_kernel(float* __restrict__ ncp, const int* __restrict__ cached_pos) {
    int i = blockIdx.x * blockDim.x + threadIdx.x;
    if (i >= L * (CC - 1)) return;
    int l = i / (CC - 1), c = i - l * (CC - 1);
    ncp[l * CC + c] = (float)cached_pos[l * CC + c + 1];
}

// Write new K/V into last cache slot of layer l, plus ncp/noff for the layer.
__global__ void cache_tail_kernel(float* __restrict__ nck, float* __restrict__ ncv,
                                  float* __restrict__ ncp, float* __restrict__ noff,
                                  const float* __restrict__ krope, const float* __restrict__ qkv,
                                  const int* __restrict__ offsets, int layer) {
    int tid = blockIdx.x * blockDim.x + threadIdx.x;
    if (tid < H * HD) {
        int h = tid >> 6, d = tid & 63;
        size_t base = (((size_t)layer * H + h) * CC + (CC - 1)) * HD + d;
        nck[base] = krope[tid];
        ncv[base] = qkv[2 * D + tid];
    }
    if (tid == 0) {
        int pq = offsets[layer];
        ncp[layer * CC + CC - 1] = (float)pq;
        noff[layer] = (float)(pq + 1);
    }
}

__global__ void misc_out_kernel(float* __restrict__ alen, float* __restrict__ toff,
                                const int* __restrict__ toff_in) {
    alen[0] = 1.0f;
    toff[0] = (float)(toff_in[0] + 1);
}

// ---------------------------------------------------------------------------
extern "C" void kernel_launch(void* const* d_in, const int* in_sizes, int n_in,
                              void* d_out, int out_size, void* d_ws, size_t ws_size,
                              hipStream_t stream) {
    const int*   codes      = (const int*)  d_in[0];
    const float* codebooks  = (const float*)d_in[1];
    const float* W_in       = (const float*)d_in[2];
    const float* b_in       = (const float*)d_in[3];
    const float* ln1_g      = (const float*)d_in[4];
    const float* ln1_b      = (const float*)d_in[5];
    const float* Wqkv       = (const float*)d_in[6];
    const float* Wo         = (const float*)d_in[7];
    const float* ls1        = (const float*)d_in[8];
    const float* ln2_g      = (const float*)d_in[9];
    const float* ln2_b      = (const float*)d_in[10];
    const float* Wff1       = (const float*)d_in[11];
    const float* bff1       = (const float*)d_in[12];
    const float* Wff2       = (const float*)d_in[13];
    const float* bff2       = (const float*)d_in[14];
    const float* ls2        = (const float*)d_in[15];
    const float* W_out      = (const float*)d_in[16];
    const float* b_out      = (const float*)d_in[17];
    const float* cached_k   = (const float*)d_in[18];
    const float* cached_v   = (const float*)d_in[19];
    const int*   cached_pos = (const int*)  d_in[20];
    const int*   offsets    = (const int*)  d_in[21];
    const int*   toff_in    = (const int*)  d_in[22];

    float* out   = (float*)d_out;
    float* audio = out;                                // 512
    float* alen  = out + QD;                           // 1
    float* nck   = out + QD + 1;                       // L*H*C*HD
    float* ncv   = nck + (size_t)L * H * CC * HD;
    float* ncp   = ncv + (size_t)L * H * CC * HD;      // L*C
    float* noff  = ncp + (size_t)L * CC;               // L
    float* toff  = noff + L;                           // 1

    float* ws      = (float*)d_ws;
    float* x       = ws;            // 1024
    float* hbuf    = ws + 1024;     // 1024
    float* qkv     = ws + 2048;     // 3072
    float* qrope   = ws + 5120;     // 1024
    float* krope   = ws + 6144;     // 1024
    float* attnout = ws + 7168;     // 1024
    float* tmp1    = ws + 8192;     // 1024 (also used pre-FF2)
    float* ffbuf   = ws + 9216;     // 4096
    float* emb     = ws + 13312;    // 512

    auto gemv = [&](const float* xin, const float* W, float* y, int K, int N) {
        int waves = (N >> 4) * (K / KCHUNK);
        int blocks = (waves * 32 + 255) / 256;
        wmma_gemv_kernel<<<blocks, 256, 0, stream>>>(xin, W, y, K, N);
    };

    // --- Cache shift (independent of layer compute; big bandwidth streams) ---
    {
        const int per4 = (CC - 1) * HD / 4;
        dim3 grid((per4 + 255) / 256, L * H);
        shift_cache_kernel<<<grid, 256, 0, stream>>>(nck, cached_k);
        shift_cache_kernel<<<grid, 256, 0, stream>>>(ncv, cached_v);
        shift_pos_kernel<<<(L * (CC - 1) + 255) / 256, 256, 0, stream>>>(ncp, cached_pos);
    }

    // --- Embedding + input projection ---
    embed_kernel<<<(QD + 255) / 256, 256, 0, stream>>>(codes, codebooks, emb);
    init_kernel<<<(D + 255) / 256, 256, 0, stream>>>(x, b_in, D);
    gemv(emb, W_in, x, QD, D);

    // --- Transformer layers (sequential) ---
    for (int l = 0; l < L; ++l) {
        const float* Wqkv_l = Wqkv + (size_t)l * D * 3 * D;
        const float* Wo_l   = Wo   + (size_t)l * D * D;
        const float* Wff1_l = Wff1 + (size_t)l * D * FF;
        const float* Wff2_l = Wff2 + (size_t)l * FF * D;

        ln_kernel<<<1, 256, 0, stream>>>(x, ln1_g + l * D, ln1_b + l * D, hbuf);
        init_kernel<<<(3 * D + 255) / 256, 256, 0, stream>>>(qkv, nullptr, 3 * D);
        gemv(hbuf, Wqkv_l, qkv, D, 3 * D);
        rope_kernel<<<2, 256, 0, stream>>>(qkv, offsets, l, qrope, krope);

        attn_kernel<<<H, 256, 0, stream>>>(qrope, krope, qkv, cached_k, cached_v,
                                           cached_pos, offsets, l, attnout);

        cache_tail_kernel<<<(H * HD + 255) / 256, 256, 0, stream>>>(
            nck, ncv, ncp, noff, krope, qkv, offsets, l);

        init_kernel<<<(D + 255) / 256, 256, 0, stream>>>(tmp1, nullptr, D);
        gemv(attnout, Wo_l, tmp1, D, D);
        resid_kernel<<<(D + 255) / 256, 256, 0, stream>>>(x, ls1 + l * D, tmp1, D);

        ln_kernel<<<1, 256, 0, stream>>>(x, ln2_g + l * D, ln2_b + l * D, hbuf);
        init_kernel<<<(FF + 255) / 256, 256, 0, stream>>>(ffbuf, bff1 + l * FF, FF);
        gemv(hbuf, Wff1_l, ffbuf, D, FF);
        gelu_kernel<<<(FF + 255) / 256, 256, 0, stream>>>(ffbuf, FF);
        init_kernel<<<(D + 255) / 256, 256, 0, stream>>>(tmp1, bff2 + l * D, D);
        gemv(ffbuf, Wff2_l, tmp1, FF, D);
        resid_kernel<<<(D + 255) / 256, 256, 0, stream>>>(x, ls2 + l * D, tmp1, D);
    }

    // --- Output projection + misc outputs ---
    init_kernel<<<(QD + 255) / 256, 256, 0, stream>>>(audio, b_out, QD);
    gemv(x, W_out, audio, D, QD);
    misc_out_kernel<<<1, 1, 0, stream>>>(alen, toff, toff_in);
}